// pAUC_KLDRO_Loss_84378927497634
// MI455X (gfx1250) — compile-verified
//
#include <hip/hip_runtime.h>

#define NMAX        16384
#define MARGIN_F    1.0f
#define BETA_F      0.1f
#define NEG_CHUNKS  8
#define STAGE       2048
#define WPB         8        // waves per 256-thread block (wave32)

typedef __attribute__((ext_vector_type(2))) float v2f;
typedef __attribute__((ext_vector_type(8))) float v8f;

// ---- workspace layout (float units) ----
// [0..3]  : int cnt[2] (P, Nn) + pad
// pos_f   : ws + 4
// pos_u   : ws + 4 +   NMAX
// neg_f   : ws + 4 + 2*NMAX
// rs_e    : ws + 4 + 3*NMAX
// rs_es   : ws + 4 + 4*NMAX

__global__ void k_zero(float* ws, float* out) {
  int i = blockIdx.x * blockDim.x + threadIdx.x;
  if (i == 0) { out[0] = 0.0f; ((int*)ws)[0] = 0; ((int*)ws)[1] = 0; }
  float* rs = ws + 4 + 3 * NMAX;
  if (i < 2 * NMAX) rs[i] = 0.0f;
}

__global__ void k_compact(const float* __restrict__ y_pred,
                          const int* __restrict__ y_true,
                          const long long* __restrict__ index_p,
                          const float* __restrict__ u_pos,
                          float* ws, int n) {
  int i = blockIdx.x * blockDim.x + threadIdx.x;
  if (i >= n) return;
  int*   cnt   = (int*)ws;
  float* pos_f = ws + 4;
  float* pos_u = pos_f + NMAX;
  float* neg_f = pos_u + NMAX;
  float v = y_pred[i];
  if (y_true[i] == 1) {
    int s = atomicAdd(&cnt[0], 1);
    pos_f[s] = v;
    long long id = index_p[i];
    pos_u[s] = u_pos[id >= 0 ? id : 0];
  } else {
    int s = atomicAdd(&cnt[1], 1);
    neg_f[s] = v;
  }
}

__device__ __forceinline__ void pair_eval(float fp, v2f fn2, v2f& ae, v2f& aes) {
#pragma unroll
  for (int k = 0; k < 2; ++k) {
    float d = MARGIN_F - (fp - fn2[k]);
    float h = d > 0.0f ? d : 0.0f;
    float s = h * h;
    float e = __expf(s);            // v_exp_f32 (LAMBDA == 1)
    ae[k]  = e;
    aes[k] = e * s;
  }
}

// Each wave owns a 16-row tile of positives; 8 waves/block share an LDS-staged
// strip of negatives (filled with global_load_async_to_lds_b32 / ASYNCcnt).
// Row sums of exp(s) and exp(s)*s accumulate via V_WMMA_F32_16X16X4_F32 with
// an all-ones B matrix: D[m,n] += sum_k A[m,k].
__global__ void __launch_bounds__(256) k_main(float* ws) {
  __shared__ float sneg[STAGE];
  int*   cnt   = (int*)ws;
  const int P  = cnt[0];
  const int Nn = cnt[1];
  float* pos_f = ws + 4;
  float* pos_u = pos_f + NMAX;
  float* neg_f = pos_u + NMAX;
  float* rs_e  = neg_f + NMAX;
  float* rs_es = rs_e  + NMAX;

  const int tid     = threadIdx.x;
  const int lane    = tid & 31;
  const int wave    = tid >> 5;
  const int rowTile = blockIdx.x * WPB + wave;
  const bool active = (rowTile * 16) < P;     // wave-uniform

  // A-matrix layout for 16x4 f32: lane holds row = lane&15;
  // lanes 0-15 carry K=0,1 ; lanes 16-31 carry K=2,3.
  const int row    = rowTile * 16 + (lane & 15);
  const int rclamp = (P > 0) ? (row < P ? row : P - 1) : 0;
  const float fp   = pos_f[rclamp];
  const int koff   = (lane >> 4) << 1;        // 0 or 2

  const int chunk  = (Nn + NEG_CHUNKS - 1) / NEG_CHUNKS;
  const int jStart = blockIdx.y * chunk;
  const int jEnd0  = jStart + chunk;
  const int jEnd   = jEnd0 < Nn ? jEnd0 : Nn;

  v8f acc_e  = {};
  v8f acc_es = {};
  const v2f ones = {1.0f, 1.0f};

  // LDS byte offset of this thread's staging slot (low 32 bits of the
  // generic pointer to LDS == LDS offset on gfx1250).
  const unsigned ldsBase = (unsigned)(size_t)(void*)&sneg[tid];

  for (int base = jStart; base < jEnd; base += STAGE) {
    const int cnts = (jEnd - base) < STAGE ? (jEnd - base) : STAGE;
    __syncthreads();
    {
      // Async copy: each pass writes global neg_f[base + tid] to LDS slot sneg[i], i = tid, tid+256, ...
      const unsigned long long gbase =
          (unsigned long long)(const void*)(neg_f + base);
      for (int i = tid; i < cnts; i += 256) {
        const unsigned voff    = (unsigned)((i - tid) * 4);  // byte offset past ldsBase (&sneg[tid])
        const unsigned ldsaddr = ldsBase + voff;
        asm volatile("global_load_async_to_lds_b32 %0, %1, %2 offset:%3"
                     :
                     : "v"(ldsaddr), "v"((unsigned)(tid * 4)), "s"(gbase),
                       "n"(0), "i"(0)
                     : "memory");
        (void)voff;
      }
      asm volatile("s_wait_asynccnt 0x0" ::: "memory");
    }
    if (base + STAGE < jEnd)
      __builtin_prefetch(&neg_f[base + STAGE + tid], 0, 0);  // global_prefetch_b8
    __syncthreads();

    if (active) {                    // wave-uniform: EXEC all-1s around WMMA
      const int nFull = cnts & ~7;
      for (int j0 = 0; j0 < nFull; j0 += 8) {
        // one ds_load_b64 per 4-wide tile
        const v2f fn0 = *(const v2f*)&sneg[j0 + koff];
        const v2f fn1 = *(const v2f*)&sneg[j0 + 4 + koff];
        v2f ae0, aes0, ae1, aes1;
        pair_eval(fp, fn0, ae0, aes0);
        pair_eval(fp, fn1, ae1, aes1);
        acc_e  = __builtin_amdgcn_wmma_f32_16x16x4_f32(
                     false, ae0,  false, ones, (short)0, acc_e,  false, false);
        acc_es = __builtin_amdgcn_wmma_f32_16x16x4_f32(
                     false, aes0, false, ones, (short)0, acc_es, false, false);
        acc_e  = __builtin_amdgcn_wmma_f32_16x16x4_f32(
                     false, ae1,  false, ones, (short)0, acc_e,  false, false);
        acc_es = __builtin_amdgcn_wmma_f32_16x16x4_f32(
                     false, aes1, false, ones, (short)0, acc_es, false, false);
      }
      for (int j0 = nFull; j0 < cnts; j0 += 4) {   // masked tail (<=2 iters)
        v2f ae, aes;
#pragma unroll
        for (int k = 0; k < 2; ++k) {
          const int jj  = j0 + koff + k;
          const bool ok = jj < cnts;
          const float fn = sneg[ok ? jj : 0];
          float d = MARGIN_F - (fp - fn);
          float h = d > 0.0f ? d : 0.0f;
          float s = h * h;
          float e = ok ? __expf(s) : 0.0f;
          ae[k]  = e;
          aes[k] = e * s;
        }
        acc_e  = __builtin_amdgcn_wmma_f32_16x16x4_f32(
                     false, ae,  false, ones, (short)0, acc_e,  false, false);
        acc_es = __builtin_amdgcn_wmma_f32_16x16x4_f32(
                     false, aes, false, ones, (short)0, acc_es, false, false);
      }
    }
  }

  // D layout: VGPR r -> (M=r, N=lane) lanes 0-15, (M=r+8, N=lane-16) lanes 16-31.
  // Columns identical (B==ones): lane 0 holds rows 0-7, lane 16 rows 8-15.
  if (active && (lane == 0 || lane == 16)) {
    const int mbase = rowTile * 16 + ((lane >> 4) << 3);
#pragma unroll
    for (int r = 0; r < 8; ++r) {
      const int rr = mbase + r;
      if (rr < P) {
        atomicAdd(&rs_e[rr],  acc_e[r]);
        atomicAdd(&rs_es[rr], acc_es[r]);
      }
    }
  }
}

__global__ void k_final(float* ws, float* out) {
  int*   cnt   = (int*)ws;
  const int P  = cnt[0];
  const int Nn = cnt[1];
  float* pos_u = ws + 4 +     NMAX;
  float* rs_e  = ws + 4 + 3 * NMAX;
  float* rs_es = ws + 4 + 4 * NMAX;
  const float invNn = (Nn > 0) ? 1.0f / (float)Nn : 0.0f;
  const float scale = (P > 0 && Nn > 0) ? 1.0f / ((float)P * (float)Nn) : 0.0f;

  float part = 0.0f;
  for (int p = blockIdx.x * blockDim.x + threadIdx.x; p < P;
       p += blockDim.x * gridDim.x) {
    const float mean_e = rs_e[p] * invNn;
    const float u_new  = (1.0f - BETA_F) * pos_u[p] + BETA_F * mean_e;
    part += rs_es[p] / u_new;
  }
  if (part != 0.0f) atomicAdd(out, part * scale);
}

extern "C" void kernel_launch(void* const* d_in, const int* in_sizes, int n_in,
                              void* d_out, int out_size, void* d_ws, size_t ws_size,
                              hipStream_t stream) {
  const float*     y_pred  = (const float*)d_in[0];
  const int*       y_true  = (const int*)d_in[1];
  const long long* index_p = (const long long*)d_in[2];
  const float*     u_pos   = (const float*)d_in[3];
  float* out = (float*)d_out;
  float* ws  = (float*)d_ws;
  const int n = in_sizes[0];   // N = 16384

  k_zero<<<(2 * NMAX + 255) / 256, 256, 0, stream>>>(ws, out);
  k_compact<<<(n + 255) / 256, 256, 0, stream>>>(y_pred, y_true, index_p, u_pos, ws, n);

  dim3 grid((NMAX / 16 + WPB - 1) / WPB, NEG_CHUNKS);  // worst-case P coverage
  k_main<<<grid, 256, 0, stream>>>(ws);

  k_final<<<32, 256, 0, stream>>>(ws, out);
}